// PreprocessTargets_9165460209903
// MI455X (gfx1250) — compile-verified
//
#include <hip/hip_runtime.h>
#include <math.h>

// ---------------------------------------------------------------------------
// PreprocessTargets for MI455X (gfx1250, wave32)
//
// Pipeline per head (F in {80,40,20}):
//   K1  score+keys : WMMA f32 16x16x4 row-sums -> 64-bit order keys (491520)
//   K2  radix-select (8x hist + scan) -> exact 16384-th largest key
//   K3  compact winners (exactly 16384, keys are unique)
//   K4  one-WGP bitonic sort of 128KB keys in LDS, fed by async global<->LDS
//   K5  decode winners -> (anchor_ind, y_ind, x_ind, t_boxes) as float
// ---------------------------------------------------------------------------

typedef unsigned int       u32;
typedef unsigned long long u64;
typedef __attribute__((ext_vector_type(2))) float v2f;
typedef __attribute__((ext_vector_type(8))) float v8f;

#define NB      98304u          // 8*4096*3 rows per head
#define NCAND   491520u         // 5 * NB candidates
#define KSEL    16384u          // top-k
#define GRIDTH  (480u * 256u)   // grid-stride for streaming kernels

// ---------------------------------------------------------------------------
// K1: per-row score via V_WMMA_F32_16X16X4_F32 (A = 16 rows x 4 vals, B = ones)
//     then mask/tap conditions and 5 key writes per row.
// key = score_bits << 32 | (0xFFFFFFFF - q)   (descending sort == jax top_k)
// ---------------------------------------------------------------------------
__global__ __launch_bounds__(256) void k_score_keys(
    const float* __restrict__ labels, const float* __restrict__ anch,
    u64* __restrict__ keys, int F)
{
    __shared__ float svals[256][8];
    __shared__ float ssum[256];

    const int tid   = threadIdx.x;
    const int row   = blockIdx.x * 256 + tid;        // 384*256 == NB exactly
    const int a     = row % 3;
    const int t2    = row / 3;
    const int nn    = t2 & 4095;
    const int batch = t2 >> 12;
    const float fF  = (float)F;

    const float* lb = labels + (batch * 4096 + nn) * 5;
    const float cls = lb[0];
    const float x   = lb[1] * fF;
    const float y   = lb[2] * fF;
    const float w   = lb[3] * fF;
    const float h   = lb[4] * fF;

    svals[tid][0] = cls;  svals[tid][1] = x;
    svals[tid][2] = y;    svals[tid][3] = w;
    svals[tid][4] = h;    svals[tid][5] = (float)(a + 1);
    svals[tid][6] = 0.f;  svals[tid][7] = 0.f;
    __syncthreads();

    // --- WMMA row sums: each wave reduces 32 rows (2 x 16-row tiles) --------
    {
        const int wv = tid >> 5, ln = tid & 31;
        const int half = ln >> 4, lm = ln & 15;     // A layout: lanes 0-15 K0/K1, 16-31 K2/K3
        v2f bones; bones.x = 1.0f; bones.y = 1.0f;  // B(4x16) all ones
        #pragma unroll
        for (int p = 0; p < 2; ++p) {
            const int R0 = wv * 32 + p * 16;
            const int rs = R0 + lm;
            v2f a1, a2;
            a1.x = svals[rs][half * 2 + 0];
            a1.y = svals[rs][half * 2 + 1];
            a2.x = svals[rs][4 + half * 2 + 0];
            a2.y = svals[rs][4 + half * 2 + 1];
            v8f c = {};
            c = __builtin_amdgcn_wmma_f32_16x16x4_f32(false, a1, false, bones,
                                                      (short)0, c, false, false);
            c = __builtin_amdgcn_wmma_f32_16x16x4_f32(false, a2, false, bones,
                                                      (short)0, c, false, false);
            // C layout: VGPR r -> M=r (lanes 0-15) / M=r+8 (lanes 16-31), any N column
            if (lm == 0) {
                const int base = R0 + half * 8;
                #pragma unroll
                for (int m2 = 0; m2 < 8; ++m2) ssum[base + m2] = c[m2];
            }
        }
    }
    __syncthreads();

    const float S = ssum[tid];

    // mask from batch-0 w/h (reference: m = max(worse, axis=3)[0])
    const float* lb0 = labels + nn * 5;
    const float w0 = lb0[3] * fF, h0 = lb0[4] * fF;
    const float aw = anch[a * 2 + 0], ah = anch[a * 2 + 1];
    const float r1 = w0 / aw, r2 = h0 / ah;
    float mm = fmaxf(fmaxf(r1, 1.0f / r1), fmaxf(r2, 1.0f / r2));
    mm = (mm != 0.0f) ? mm : 65504.0f;
    const bool mask = mm < 4.0f;

    const float xb = mask ? x : 0.0f;
    const float yb = mask ? y : 0.0f;
    const bool c1 = (fmodf(xb, 1.0f) < 0.5f) && (xb > 1.0f);
    const bool c2 = (fmodf(yb, 1.0f) < 0.5f) && (yb > 1.0f);
    const float ix = (xb != 0.0f) ? (fF - xb) : 0.0f;
    const float iy = (yb != 0.0f) ? (fF - yb) : 0.0f;
    const bool c3 = (fmodf(ix, 1.0f) < 0.5f) && (ix > 1.0f);
    const bool c4 = (fmodf(iy, 1.0f) < 0.5f) && (iy > 1.0f);

    const u32 sb = __float_as_uint(S);               // score >= 1 when selected
    const bool sel5[5] = { mask, c1, c2, c3, c4 };   // taps 1-4 imply mask
    const u32 q0 = ((u32)batch * 4096u + (u32)nn) * 3u + (u32)a;
    #pragma unroll
    for (u32 t = 0; t < 5u; ++t) {
        const u32 q = q0 + t * NB;
        const u32 s = sel5[t] ? sb : 0u;
        keys[q] = ((u64)s << 32) | (u64)(0xFFFFFFFFu - q);
    }
}

// ---------------------------------------------------------------------------
// state layout (u32): [0]=compact counter  [1]=prefix lo  [2]=prefix hi
//                     [3]=k-remaining      [8..263]=hist256
// ---------------------------------------------------------------------------
__global__ void k_sel_init(u32* __restrict__ state)
{
    state[8 + threadIdx.x] = 0u;
    if (threadIdx.x == 0) {
        state[0] = 0u; state[1] = 0u; state[2] = 0u; state[3] = KSEL;
    }
}

__global__ __launch_bounds__(256) void k_hist(
    const u64* __restrict__ keys, u32* __restrict__ state, int p)
{
    __shared__ u32 sh[256];
    sh[threadIdx.x] = 0u;
    __syncthreads();

    const u64 prefix = ((u64)state[2] << 32) | (u64)state[1];
    const u64 pmask  = (p >= 7) ? 0ull : (~0ull << (u32)((p + 1) * 8));
    const int shift  = p * 8;

    for (u32 i = blockIdx.x * 256u + threadIdx.x; i < NCAND; i += GRIDTH) {
        __builtin_prefetch(&keys[i + GRIDTH], 0, 1);   // -> global_prefetch_b8
        const u64 k = keys[i];
        if (((k ^ prefix) & pmask) == 0ull)
            atomicAdd(&sh[(u32)(k >> shift) & 255u], 1u);
    }
    __syncthreads();
    const u32 c = sh[threadIdx.x];
    if (c) atomicAdd(&state[8 + threadIdx.x], c);
}

__global__ void k_scan(u32* __restrict__ state, int p)
{
    if (threadIdx.x == 0) {
        u32 krem = state[3];
        u32 cum  = 0u;
        u32 bin  = 0u;
        for (int b2 = 255; b2 >= 0; --b2) {
            const u32 c = state[8 + b2];
            if (cum + c >= krem) { bin = (u32)b2; krem -= cum; break; }
            cum += c;
        }
        u64 prefix = ((u64)state[2] << 32) | (u64)state[1];
        prefix |= ((u64)bin) << (p * 8);
        state[1] = (u32)prefix;
        state[2] = (u32)(prefix >> 32);
        state[3] = krem;
    }
    __syncthreads();
    state[8 + threadIdx.x] = 0u;   // clear hist for next pass
}

__global__ __launch_bounds__(256) void k_compact(
    const u64* __restrict__ keys, u32* __restrict__ state, u64* __restrict__ sel)
{
    const u64 T = ((u64)state[2] << 32) | (u64)state[1];  // exact KSEL-th largest
    for (u32 i = blockIdx.x * 256u + threadIdx.x; i < NCAND; i += GRIDTH) {
        const u64 k = keys[i];
        if (k >= T) {
            const u32 pos = atomicAdd(&state[0], 1u);
            if (pos < KSEL) sel[pos] = k;                 // exactly KSEL winners
        }
    }
}

// ---------------------------------------------------------------------------
// K4: single-workgroup bitonic sort (descending) of 16384 keys in 128KB LDS.
//     Bulk global<->LDS via gfx1250 async ops (ASYNCcnt path, no VGPR bounce).
// ---------------------------------------------------------------------------
__global__ __launch_bounds__(1024) void k_sort(u64* __restrict__ sel)
{
    extern __shared__ u64 skey[];   // 16384 * 8B = 128KB (<= 320KB/WGP)
    const u32 tid = threadIdx.x;

    #pragma unroll
    for (int i = 0; i < 16; ++i) {
        const u32 idx = (u32)i * 1024u + tid;
        const u32 lds = (u32)(size_t)(&skey[idx]);      // low 32 bits = LDS byte addr
        const u32 off = idx * 8u;
        asm volatile("global_load_async_to_lds_b64 %0, %1, %2"
                     :: "v"(lds), "v"(off), "s"(sel) : "memory");
    }
    asm volatile("s_wait_asynccnt 0" ::: "memory");
    __syncthreads();

    for (u32 k = 2u; k <= KSEL; k <<= 1) {
        for (u32 j = k >> 1; j > 0u; j >>= 1) {
            #pragma unroll 1
            for (u32 i = tid; i < KSEL; i += 1024u) {
                const u32 l = i ^ j;
                if (l > i) {
                    const u64 va = skey[i], vb = skey[l];
                    const bool up = ((i & k) == 0u);
                    if (up ? (va < vb) : (va > vb)) { skey[i] = vb; skey[l] = va; }
                }
            }
            __syncthreads();
        }
    }

    #pragma unroll
    for (int i = 0; i < 16; ++i) {
        const u32 idx = (u32)i * 1024u + tid;
        const u32 lds = (u32)(size_t)(&skey[idx]);
        const u32 off = idx * 8u;
        asm volatile("global_store_async_from_lds_b64 %0, %1, %2"
                     :: "v"(off), "v"(lds), "s"(sel) : "memory");
    }
    asm volatile("s_wait_asynccnt 0" ::: "memory");
}

// ---------------------------------------------------------------------------
// K5: decode sorted winners -> outputs (all as float in d_out)
// per head layout: anchor[16384] | y[16384] | x[16384] | boxes[16384*5]
// ---------------------------------------------------------------------------
__global__ __launch_bounds__(256) void k_decode(
    const u64* __restrict__ sel, const float* __restrict__ labels,
    float* __restrict__ out, int F, int head)
{
    const u32 r = blockIdx.x * 256u + threadIdx.x;    // 64*256 == KSEL
    const u64 key = sel[r];
    const u32 q     = 0xFFFFFFFFu - (u32)(key & 0xFFFFFFFFull);
    const u32 sbits = (u32)(key >> 32);

    float anchor_f = 0.f, yi = 0.f, xi = 0.f;
    float bx0 = 0.f, bx1 = 0.f, bx2 = 0.f, bx3 = 0.f, bx4 = 0.f;

    if (sbits != 0u) {                     // zero-score rows decode to all zeros
        const u32 t  = q / NB;
        const u32 rr = q % NB;
        const u32 a  = rr % 3u;
        const u32 t2 = rr / 3u;
        const u32 nn = t2 & 4095u;
        const u32 batch = t2 >> 12;
        const float fF = (float)F;
        const float* lb = labels + (batch * 4096u + nn) * 5u;
        const float cls = lb[0];
        const float x = lb[1] * fF, y = lb[2] * fF;
        const float w = lb[3] * fF, h = lb[4] * fF;
        const float ox = (t == 1u) ? 0.5f : ((t == 3u) ? -0.5f : 0.0f);
        const float oy = (t == 2u) ? 0.5f : ((t == 4u) ? -0.5f : 0.0f);
        int ixi = (x != 0.0f) ? (int)(x - ox) : 0;    // trunc == astype(int32)
        int iyi = (y != 0.0f) ? (int)(y - oy) : 0;
        ixi = min(max(ixi, 0), F - 1);
        iyi = min(max(iyi, 0), F - 1);
        anchor_f = (float)a;                          // clip((a+1)-1, 0, 4)
        xi = (float)ixi;  yi = (float)iyi;
        bx0 = cls;
        bx1 = x - (float)ixi;
        bx2 = y - (float)iyi;
        bx3 = w;
        bx4 = h;
    }

    float* ob = out + (size_t)head * (8u * KSEL);
    ob[r]                 = anchor_f;
    ob[KSEL + r]          = yi;
    ob[2u * KSEL + r]     = xi;
    float* tb = ob + 3u * KSEL + (size_t)r * 5u;
    tb[0] = bx0; tb[1] = bx1; tb[2] = bx2; tb[3] = bx3; tb[4] = bx4;
}

// ---------------------------------------------------------------------------
extern "C" void kernel_launch(void* const* d_in, const int* in_sizes, int n_in,
                              void* d_out, int out_size, void* d_ws, size_t ws_size,
                              hipStream_t stream)
{
    const float* labels = (const float*)d_in[0];
    float* out = (float*)d_out;

    // workspace: keys 3,932,160B | sel 131,072B | state 1,056B  (~4.1MB total)
    char* ws   = (char*)d_ws;
    u64*  keys = (u64*)ws;
    u64*  sel  = (u64*)(ws + 3932160);
    u32*  state = (u32*)(ws + 3932160 + 131072);

    const int Fs[3] = { 80, 40, 20 };
    for (int head = 0; head < 3; ++head) {
        const float* anch = (const float*)d_in[1 + head];
        k_score_keys<<<384, 256, 0, stream>>>(labels, anch, keys, Fs[head]);
        k_sel_init<<<1, 256, 0, stream>>>(state);
        for (int p = 7; p >= 0; --p) {
            k_hist<<<480, 256, 0, stream>>>(keys, state, p);
            k_scan<<<1, 256, 0, stream>>>(state, p);
        }
        k_compact<<<480, 256, 0, stream>>>(keys, state, sel);
        k_sort<<<1, 1024, 131072, stream>>>(sel);
        k_decode<<<64, 256, 0, stream>>>(sel, labels, out, Fs[head], head);
    }
}